// Model_8306466751104
// MI455X (gfx1250) — compile-verified
//
#include <hip/hip_runtime.h>
#include <hip/hip_bf16.h>
#include <math.h>

typedef __attribute__((ext_vector_type(2))) float v2f;
typedef __attribute__((ext_vector_type(8))) float v8f;

#define F_FACES 2048
#define IMG_H 192
#define IMG_W 192
#define EPS_MISS 1e-7f
#define INV_SIGMA 100.0f          // 1 / 0.01
#define TAN_VIEW 0.57735026919f   // tan(30 deg)
#define LOG2E 1.44269504089f
#define P_CLAMP 1e9f              // inside < 1e-9 is indistinguishable vs EPS

// ---------------------------------------------------------------------------
// Kernel 0: zero the scalar output (harness poisons d_out with 0xAA).
// ---------------------------------------------------------------------------
__global__ void zero_out(float* out) { out[0] = 0.0f; }

// ---------------------------------------------------------------------------
// Kernel 1: per-face-edge coefficient prep.
// For face f, edge i the WMMA row is [A, B, C, 0] with the full scale
//   -(sgn/elen) * INV_SIGMA * LOG2E
// folded in, so the WMMA output is directly the exp2 argument:
//   exp2(D) == exp(-d/SIGMA)  ->  sigmoid(d/SIGMA) = 1/(1+exp2(D)).
// ---------------------------------------------------------------------------
__global__ void prep_faces(const float* __restrict__ verts,
                           const int* __restrict__ faces,
                           const float* __restrict__ cam,
                           float* __restrict__ coef) {
  int f = blockIdx.x * blockDim.x + threadIdx.x;
  if (f >= F_FACES) return;

  float ex = cam[0], ey = cam[1], ez = cam[2];
  // z_axis = normalize(-eye)
  float zl = fmaxf(sqrtf(ex * ex + ey * ey + ez * ez), 1e-8f);
  float zx = -ex / zl, zy = -ey / zl, zz = -ez / zl;
  // x_axis = normalize(cross([0,1,0], z)) = normalize((zz, 0, -zx))
  float xx = zz, xz = -zx;
  float xl = fmaxf(sqrtf(xx * xx + xz * xz), 1e-8f);
  xx /= xl; xz /= xl;
  // y_axis = cross(z, x)
  float yx = zy * xz;
  float yy = zz * xx - zx * xz;
  float yz = -zy * xx;

  float pxv[3], pyv[3];
#pragma unroll
  for (int i = 0; i < 3; ++i) {
    int vi = faces[f * 3 + i];
    float vx = verts[vi * 3 + 0] - ex;
    float vy = verts[vi * 3 + 1] - ey;
    float vz = verts[vi * 3 + 2] - ez;
    float cx = xx * vx + xz * vz;
    float cy = yx * vx + yy * vy + yz * vz;
    float cz = zx * vx + zy * vy + zz * vz;
    cz = fmaxf(cz, 1e-5f);
    float inv = 1.0f / (cz * TAN_VIEW);
    pxv[i] = cx * inv;
    pyv[i] = cy * inv;
  }

  float e01x = pxv[1] - pxv[0], e01y = pyv[1] - pyv[0];
  float e02x = pxv[2] - pxv[0], e02y = pyv[2] - pyv[0];
  float area2 = e01x * e02y - e01y * e02x;
  float sgn = (area2 >= 0.0f) ? 1.0f : -1.0f;

#pragma unroll
  for (int i = 0; i < 3; ++i) {
    int j = (i + 1) % 3;  // edges = fv[[1,2,0]] - fv
    float eex = pxv[j] - pxv[i];
    float eey = pyv[j] - pyv[i];
    float c = eey * pxv[i] - eex * pyv[i];
    float elen = fmaxf(sqrtf(eex * eex + eey * eey), 1e-8f);
    // negative scale: WMMA result is -d/SIGMA in log2-of-e units
    float s = -(sgn / elen) * INV_SIGMA * LOG2E;
    float4 o;
    o.x = eex * s;   // A  (multiplies py)
    o.y = -eey * s;  // B  (multiplies px)
    o.z = c * s;     // C
    o.w = 0.0f;      // K pad
    reinterpret_cast<float4*>(coef)[i * F_FACES + f] = o;
  }
}

// ---------------------------------------------------------------------------
// Kernel 2: main loss. One wave32 per 16-pixel strip (fixed y, 16 x's).
// D[face, pixel] = WMMA_F32_16x16x4( coeffs[16 faces x 4], [py px 1 0]^T x 16 px )
// Three WMMAs (one per edge) per 16-face tile. Per row:
//   P    = (1+exp2(D0))(1+exp2(D1))(1+exp2(D2)), clamped to 1e9
//   miss = 1 + EPS - 1/P = (P*(1+EPS) - 1) / P
// Batched over groups of 4 rows:
//   acc += log2(Π num) - log2(Π P)
// -> 24x v_exp_f32 + 4x v_log_f32 per tile; no rcp, no div, no base-
// conversion multiplies. Final: image = 1 - exp2(acc).
// ---------------------------------------------------------------------------
__global__ void __launch_bounds__(256)
silhouette_loss(const float* __restrict__ coef,
                const float* __restrict__ image_ref,
                const float* __restrict__ cam,
                float* __restrict__ out) {
  const int lane = threadIdx.x & 31;
  const int wave = threadIdx.x >> 5;
  const int strip = blockIdx.x * 8 + wave;        // 0 .. 2303 (exact fit)
  const int tiles_x = IMG_W / 16;                 // 12
  const int y = strip / tiles_x;
  const int x0 = (strip % tiles_x) * 16;
  const int n = lane & 15;                        // pixel column within tile

  const float py = 1.0f - ((float)y + 0.5f) / (float)IMG_H * 2.0f;
  const float px = ((float)(x0 + n) + 0.5f) / (float)IMG_W * 2.0f - 1.0f;

  // B operand (4x16 f32): VGPR0 = K rows {0,2}, VGPR1 = K rows {1,3}
  // lanes 0-15: (K0,K1) = (py, px); lanes 16-31: (K2,K3) = (1, 0)
  v2f b;
  if (lane < 16) { b.x = py;   b.y = px;   }
  else           { b.x = 1.0f; b.y = 0.0f; }

  const int row = lane & 15;          // M row within face tile
  const int k0 = (lane >> 4) << 1;    // 0 (lanes 0-15) or 2 (lanes 16-31)

  float acc = 0.0f;                   // sum of log2(miss)

  for (int fb = 0; fb < F_FACES; fb += 16) {
    int r = fb + row;
    v2f a0 = *reinterpret_cast<const v2f*>(coef + (size_t)(0 * F_FACES + r) * 4 + k0);
    v2f a1 = *reinterpret_cast<const v2f*>(coef + (size_t)(1 * F_FACES + r) * 4 + k0);
    v2f a2 = *reinterpret_cast<const v2f*>(coef + (size_t)(2 * F_FACES + r) * 4 + k0);

    v8f z = {};
    v8f d0 = __builtin_amdgcn_wmma_f32_16x16x4_f32(false, a0, false, b, (short)0, z, false, false);
    v8f d1 = __builtin_amdgcn_wmma_f32_16x16x4_f32(false, a1, false, b, (short)0, z, false, false);
    v8f d2 = __builtin_amdgcn_wmma_f32_16x16x4_f32(false, a2, false, b, (short)0, z, false, false);

    // Two groups of 4 rows; products stay in fp32 range:
    //   P in [1, 1e9]  -> ΠP  in [1, 1e36]
    //   num in [EPS, ~1e9] -> Πnum in [1e-28, 1e36]
    float np[2] = {1.0f, 1.0f};   // Π (P*(1+EPS) - 1)
    float dp[2] = {1.0f, 1.0f};   // Π P
#pragma unroll
    for (int rr = 0; rr < 8; ++rr) {
      float e0 = __builtin_amdgcn_exp2f(d0[rr]);   // exp(-d0/SIGMA)
      float e1 = __builtin_amdgcn_exp2f(d1[rr]);
      float e2 = __builtin_amdgcn_exp2f(d2[rr]);
      float P = (1.0f + e0) * (1.0f + e1) * (1.0f + e2);
      P = fminf(P, P_CLAMP);
      float num = fmaf(P, 1.0f + EPS_MISS, -1.0f); // P*(1+EPS) - 1 >= EPS
      const int g = rr >> 2;
      np[g] *= num;
      dp[g] *= P;
    }
    acc += __builtin_amdgcn_logf(np[0]) - __builtin_amdgcn_logf(dp[0]);
    acc += __builtin_amdgcn_logf(np[1]) - __builtin_amdgcn_logf(dp[1]);
  }

  // Each half-wave accumulated a disjoint set of 8-face rows for the same
  // pixel column; combine lane L with lane L^16.
  acc += __shfl_xor(acc, 16, 32);

  float sq = 0.0f;
  if (lane < 16) {
    // log_miss (natural) = acc * ln2; image = 1 - exp(log_miss) = 1 - exp2(acc)
    float image = 1.0f - __builtin_amdgcn_exp2f(acc);
    float diff = image_ref[y * IMG_W + x0 + n] - image;
    sq = diff * diff;
  }
  // Reduce the 16 active pixel lanes (upper half contributes 0).
  sq += __shfl_xor(sq, 8, 32);
  sq += __shfl_xor(sq, 4, 32);
  sq += __shfl_xor(sq, 2, 32);
  sq += __shfl_xor(sq, 1, 32);

  if (lane == 0) {
    float cx = cam[0], cy = cam[1], cz = cam[2];
    float dist = sqrtf(cx * cx + cy * cy + cz * cz);
    float factor = 1.0f + fmaxf(0.0f, 6.0f - dist);
    atomicAdd(out, sq * factor);
  }
}

// ---------------------------------------------------------------------------
extern "C" void kernel_launch(void* const* d_in, const int* in_sizes, int n_in,
                              void* d_out, int out_size, void* d_ws, size_t ws_size,
                              hipStream_t stream) {
  const float* vertices  = (const float*)d_in[0];   // (1, 1024, 3) f32
  const int*   faces     = (const int*)d_in[1];     // (1, 2048, 3) i32
  const float* image_ref = (const float*)d_in[2];   // (192, 192) f32
  const float* cam       = (const float*)d_in[3];   // (3,) f32
  float* out = (float*)d_out;
  float* coef = (float*)d_ws;  // 3 * 2048 * 4 floats = 96 KiB

  zero_out<<<1, 1, 0, stream>>>(out);
  prep_faces<<<F_FACES / 256, 256, 0, stream>>>(vertices, faces, cam, coef);

  // 192*192 pixels / 16 per wave = 2304 waves; 8 waves per 256-thread block.
  silhouette_loss<<<2304 / 8, 256, 0, stream>>>(coef, image_ref, cam, out);
}